// Augment_35751307772251
// MI455X (gfx1250) — compile-verified
//
#include <hip/hip_runtime.h>
#include <math.h>

#define BS 32
#define HID 1024
#define NS 8192
#define MD 128
#define NR 4
#define GAMMAF 0.95f

typedef __attribute__((ext_vector_type(2))) float v2f;
typedef __attribute__((ext_vector_type(8))) float v8f;

// ---------------- K0: sigmoid(alpha), zero norm accumulators + read_vec ----
__global__ void k0_init(const float* __restrict__ alpha, float* __restrict__ ws_f,
                        float* __restrict__ rv) {
    int t = blockIdx.x * blockDim.x + threadIdx.x;      // 64*256 = 16384
    if (t < BS * NR) ws_f[16 + t] = 1.0f / (1.0f + __expf(-alpha[t]));
    if (t < 2) ws_f[t] = 0.0f;                          // qss, mss
    if (t < BS * NR * MD) rv[t] = 0.0f;                 // read_vec accumulators
}

// ---------------- K1: 4 smallest-usage slot indices per batch (stable) -----
__global__ void k1_argmin4(const float* __restrict__ usage, int* __restrict__ lidx) {
    int b = blockIdx.x;
    int lane = threadIdx.x;                             // one wave32 per batch
    const float* u = usage + b * NS;
    int chosen[NR];
    for (int t = 0; t < NR; ++t) {
        float bv = 3.4e38f; int bi = NS;
        for (int i = lane; i < NS; i += 32) {
            bool skip = false;
            for (int s = 0; s < t; ++s) skip |= (chosen[s] == i);
            if (!skip) {
                float v = u[i];
                if (v < bv || (v == bv && i < bi)) { bv = v; bi = i; }
            }
        }
        for (int off = 16; off; off >>= 1) {
            float ov = __shfl_xor(bv, off);
            int   oi = __shfl_xor(bi, off);
            if (ov < bv || (ov == bv && oi < bi)) { bv = ov; bi = oi; }
        }
        chosen[t] = bi;                                 // all lanes agree
    }
    if (lane == 0)
        for (int t = 0; t < NR; ++t) lidx[b * NR + t] = chosen[t];
}

// ---------------- K2: query = hidden @ W + b ; accumulate ||q||^2 ----------
__global__ void k2_query(const float* __restrict__ hidden, const float* __restrict__ W,
                         const float* __restrict__ bias, float* __restrict__ q,
                         float* __restrict__ qss) {
    int t = blockIdx.x * blockDim.x + threadIdx.x;      // 16384 outputs
    int b = t >> 9, c = t & 511;
    const float* h = hidden + b * HID;
    float acc = bias[c];
    for (int i = 0; i < HID; ++i) acc = fmaf(h[i], W[i * (NR * MD) + c], acc);
    q[t] = acc;
    __shared__ float sm[256];
    sm[threadIdx.x] = acc * acc;
    __syncthreads();
    for (int s = 128; s; s >>= 1) {
        if (threadIdx.x < s) sm[threadIdx.x] += sm[threadIdx.x + s];
        __syncthreads();
    }
    if (threadIdx.x == 0) atomicAdd(qss, sm[0]);
}

// ---------------- K3: memory += ww^T q (WMMA K=4), fused scores + ||mem||^2 -
// One wave per (batch, 16-slot tile). A=ww[16n x 4r], B=q[4r x 16m], C/D=mem tile.
// mem_in is streamed with NT hint (dead after this pass); mem_out stays RT so
// the 128MB updated tensor remains resident in the 192MB L2 for k7's re-read.
__global__ void __launch_bounds__(128)
k3_update(const float* __restrict__ mem_in, const float* __restrict__ rw_in,
          const float* __restrict__ ws_alpha, const int* __restrict__ lidx,
          const float* __restrict__ q, float* __restrict__ mem_out,
          float* __restrict__ scores, float* __restrict__ mss) {
    int lane = threadIdx.x & 31;
    int w = blockIdx.x * 4 + (threadIdx.x >> 5);        // 16384 waves
    int b = w >> 9, nt = w & 511;
    int n0 = nt * 16;
    int half = lane >> 4, l = lane & 15;

    // A operand: write-weight, rows n=n0+l, K = r in {2*half, 2*half+1}
    int r0 = 2 * half;
    int n = n0 + l;
    float al0 = ws_alpha[b * NR + r0], al1 = ws_alpha[b * NR + r0 + 1];
    v2f A;
    A.x = al0 * rw_in[(b * NR + r0) * NS + n] + ((lidx[b * NR + r0] == n) ? (1.0f - al0) : 0.0f);
    A.y = al1 * rw_in[(b * NR + r0 + 1) * NS + n] + ((lidx[b * NR + r0 + 1] == n) ? (1.0f - al1) : 0.0f);

    // per-lane base pointers: all tile elements are base + constant offsets
    size_t mbase = (size_t)b * NS * MD;
    const float* cin  = mem_in  + mbase + (size_t)(n0 + 8 * half) * MD + l;
    float*       cout = mem_out + mbase + (size_t)(n0 + 8 * half) * MD + l;
    const float* qbl  = q + b * (NR * MD) + l;

    float sacc[8][4];
#pragma unroll
    for (int j = 0; j < 8; ++j)
#pragma unroll
        for (int r = 0; r < 4; ++r) sacc[j][r] = 0.0f;
    float ssq = 0.0f;

#pragma unroll
    for (int mt = 0; mt < 8; ++mt) {
        const int mo = mt * 16;
        float q0 = qbl[0 * MD + mo], q1 = qbl[1 * MD + mo];
        float q2 = qbl[2 * MD + mo], q3 = qbl[3 * MD + mo];
        v2f B; B.x = half ? q2 : q0; B.y = half ? q3 : q1;

        v8f C;
#pragma unroll
        for (int j = 0; j < 8; ++j)
            C[j] = __builtin_nontemporal_load(&cin[j * MD + mo]);

        v8f D = __builtin_amdgcn_wmma_f32_16x16x4_f32(false, A, false, B,
                                                      (short)0, C, false, false);
#pragma unroll
        for (int j = 0; j < 8; ++j)
            cout[j * MD + mo] = D[j];

#pragma unroll
        for (int j = 0; j < 8; ++j) {
            float dv = D[j];
            ssq = fmaf(dv, dv, ssq);
            sacc[j][0] = fmaf(dv, q0, sacc[j][0]);
            sacc[j][1] = fmaf(dv, q1, sacc[j][1]);
            sacc[j][2] = fmaf(dv, q2, sacc[j][2]);
            sacc[j][3] = fmaf(dv, q3, sacc[j][3]);
        }
    }

    // reduce over the 16 columns held by this lane-half
#pragma unroll
    for (int j = 0; j < 8; ++j)
#pragma unroll
        for (int r = 0; r < 4; ++r) {
            float v = sacc[j][r];
            v += __shfl_xor(v, 1); v += __shfl_xor(v, 2);
            v += __shfl_xor(v, 4); v += __shfl_xor(v, 8);
            sacc[j][r] = v;
        }
#pragma unroll
    for (int j = 0; j < 8; ++j) {
        if (l == j) {
            int nn = n0 + j + 8 * half;
            scores[(b * NR + 0) * NS + nn] = sacc[j][0];
            scores[(b * NR + 1) * NS + nn] = sacc[j][1];
            scores[(b * NR + 2) * NS + nn] = sacc[j][2];
            scores[(b * NR + 3) * NS + nn] = sacc[j][3];
        }
    }

    __shared__ float sm[128];
    sm[threadIdx.x] = ssq;
    __syncthreads();
    for (int s = 64; s; s >>= 1) {
        if (threadIdx.x < s) sm[threadIdx.x] += sm[threadIdx.x + s];
        __syncthreads();
    }
    if (threadIdx.x == 0) atomicAdd(mss, sm[0]);
}

// ---------------- K5: softmax over N with global-norm temperature ----------
__global__ void k5_softmax(const float* __restrict__ scores, const float* __restrict__ ws_f,
                           float* __restrict__ rw_out) {
    int row = blockIdx.x;                               // (b*4 + r), 128 rows
    int tid = threadIdx.x;                              // 256 threads, 32 elems each
    float inv = 1.0f / (sqrtf(ws_f[0]) * sqrtf(ws_f[1]));
    const float* s = scores + (size_t)row * NS;
    float v[32];
    float mx = -3.4e38f;
#pragma unroll
    for (int i = 0; i < 32; ++i) { v[i] = s[tid + i * 256] * inv; mx = fmaxf(mx, v[i]); }
    __shared__ float sm[256];
    sm[tid] = mx; __syncthreads();
    for (int st = 128; st; st >>= 1) {
        if (tid < st) sm[tid] = fmaxf(sm[tid], sm[tid + st]);
        __syncthreads();
    }
    mx = sm[0]; __syncthreads();
    float sum = 0.0f;
#pragma unroll
    for (int i = 0; i < 32; ++i) { v[i] = __expf(v[i] - mx); sum += v[i]; }
    sm[tid] = sum; __syncthreads();
    for (int st = 128; st; st >>= 1) {
        if (tid < st) sm[tid] += sm[tid + st];
        __syncthreads();
    }
    float invsum = 1.0f / sm[0];
    float* o = rw_out + (size_t)row * NS;
#pragma unroll
    for (int i = 0; i < 32; ++i) o[tid + i * 256] = v[i] * invsum;
}

// ---------------- K6: usage = gamma*u + sum_r rw_new + sum_r ww ------------
__global__ void k6_usage(const float* __restrict__ usage_in, const float* __restrict__ rw_in,
                         const float* __restrict__ rw_new, const float* __restrict__ ws_alpha,
                         const int* __restrict__ lidx, float* __restrict__ us_out) {
    int t = blockIdx.x * blockDim.x + threadIdx.x;      // 262144
    int b = t >> 13, n = t & 8191;
    float u = GAMMAF * usage_in[t];
#pragma unroll
    for (int r = 0; r < NR; ++r) {
        float al = ws_alpha[b * NR + r];
        u += rw_new[(b * NR + r) * NS + n];
        u += al * rw_in[(b * NR + r) * NS + n];
        if (lidx[b * NR + r] == n) u += (1.0f - al);
    }
    us_out[t] = u;
}

// ---------------- K7: read_vec = rw_new @ mem, WMMA K=4 chained accum ------
// One wave per (b, m-tile, n-chunk). A=rw rows (0-3 valid of 16), B=mem[4n x 16m].
// Inner loop unrolled by 8 so all addresses are base + immediate offsets.
__global__ void __launch_bounds__(128)
k7_readvec(const float* __restrict__ mem, const float* __restrict__ rw_new,
           float* __restrict__ rv) {
    int lane = threadIdx.x & 31;
    int w = blockIdx.x * 4 + (threadIdx.x >> 5);        // 2048 waves
    int b = w >> 6;
    int rem = w & 63;
    int mt = rem >> 3, nc = rem & 7;
    int half = lane >> 4, l = lane & 15;
    int m = mt * 16 + l;
    int nbase = nc * 1024;                              // 1024 slots per chunk

    const float* pA = rw_new + (size_t)(b * NR + (l & 3)) * NS + nbase + 2 * half;
    const float* pB = mem + (size_t)b * NS * MD + (size_t)(nbase + 2 * half) * MD + m;

    v8f C = {0.f, 0.f, 0.f, 0.f, 0.f, 0.f, 0.f, 0.f};
    for (int ko = 0; ko < 32; ++ko) {                   // 32 * 8 * K4 = 1024 slots
#pragma unroll
        for (int ki = 0; ki < 8; ++ki) {
            v2f A = *(const v2f*)(pA + ki * 4);         // contiguous pair, 8B aligned
            v2f B; B.x = pB[ki * 4 * MD]; B.y = pB[ki * 4 * MD + MD];
            C = __builtin_amdgcn_wmma_f32_16x16x4_f32(false, A, false, B,
                                                      (short)0, C, false, false);
        }
        pA += 32;
        pB += 32 * MD;
    }

    if (half == 0) {                                    // rows 0..3 = valid r
#pragma unroll
        for (int j = 0; j < 4; ++j)
            atomicAdd(&rv[b * (NR * MD) + j * MD + m], C[j]);
    }
}

// ---------------- host launch ----------------------------------------------
extern "C" void kernel_launch(void* const* d_in, const int* in_sizes, int n_in,
                              void* d_out, int out_size, void* d_ws, size_t ws_size,
                              hipStream_t stream) {
    (void)in_sizes; (void)n_in; (void)out_size; (void)ws_size;
    const float* mem_in   = (const float*)d_in[0];
    const float* hidden   = (const float*)d_in[1];
    const float* rw_in    = (const float*)d_in[2];
    const float* usage_in = (const float*)d_in[3];
    const float* alpha    = (const float*)d_in[4];
    const float* W        = (const float*)d_in[5];
    const float* bias     = (const float*)d_in[6];

    float* out     = (float*)d_out;
    float* rv      = out;                       // [32, 512]
    float* mem_out = out + 16384;               // [32, 8192, 128]
    float* rw_out  = mem_out + 33554432;        // [32, 4, 8192]
    float* us_out  = rw_out + 1048576;          // [32, 8192]

    float* ws_f     = (float*)d_ws;
    float* qss      = ws_f;                     // ||q||^2
    float* mss      = ws_f + 1;                 // ||mem_new||^2
    float* ws_alpha = ws_f + 16;                // sigmoid(alpha) [128]
    int*   lidx     = (int*)(ws_f + 160);       // least-used idx [128]
    float* qbuf     = ws_f + 512;               // query [32,4,128]
    float* scores   = ws_f + 32768;             // unscaled scores [32,4,8192]

    k0_init   <<<64,   256, 0, stream>>>(alpha, ws_f, rv);
    k1_argmin4<<<32,    32, 0, stream>>>(usage_in, lidx);
    k2_query  <<<64,   256, 0, stream>>>(hidden, W, bias, qbuf, qss);
    k3_update <<<4096, 128, 0, stream>>>(mem_in, rw_in, ws_alpha, lidx, qbuf,
                                         mem_out, scores, mss);
    k5_softmax<<<128,  256, 0, stream>>>(scores, ws_f, rw_out);
    k6_usage  <<<1024, 256, 0, stream>>>(usage_in, rw_in, rw_out, ws_alpha, lidx, us_out);
    k7_readvec<<<512,  128, 0, stream>>>(mem_out, rw_out, rv);
}